// H2KT_80101140070862
// MI455X (gfx1250) — compile-verified
//
#include <hip/hip_runtime.h>

typedef __attribute__((ext_vector_type(2))) float v2f;
typedef __attribute__((ext_vector_type(8))) float v8f;

#define DDIM 128
#define KSK 8
#define STRIDE 132              // padded row stride (dwords) -> conflict-free b64 LDS reads
#define WAVES 4
#define TOK_PER_WAVE 16
#define TOK_PER_WG (WAVES * TOK_PER_WAVE)

__launch_bounds__(128, 1)
__global__ void h2kt_fused_wmma(const int* __restrict__ q,
                                const int* __restrict__ ans,
                                const int* __restrict__ next_q,
                                const float* __restrict__ ques_table,
                                const float* __restrict__ skill_table,
                                const float* __restrict__ diff_table,
                                const float* __restrict__ area_table,
                                const int* __restrict__ ques2skill,
                                const int* __restrict__ ques2diff,
                                const int* __restrict__ ques2area,
                                const float* __restrict__ Wq,
                                const float* __restrict__ bq,
                                float* __restrict__ out,
                                int n_tokens, int pad_skill)
{
    extern __shared__ float lds[];
    float* lds_wq = lds;                                        // [128][STRIDE]
    const int tid  = threadIdx.x;
    const int wave = tid >> 5;
    const int lane = tid & 31;
    float* stag = lds + DDIM * STRIDE + wave * (TOK_PER_WAVE * STRIDE); // [16][STRIDE]

    // ---- stage Wq into LDS (row-major, padded) ----
    for (int i = tid; i < DDIM * (DDIM / 4); i += blockDim.x) {
        const int row = i >> 5;            // 32 float4 per row
        const int c4  = (i & 31) << 2;
        const float4 v = *(const float4*)(Wq + (long)row * DDIM + c4);
        float* p = lds_wq + row * STRIDE + c4;
        p[0] = v.x; p[1] = v.y; p[2] = v.z; p[3] = v.w;
    }
    __syncthreads();

    const int tokBase = blockIdx.x * TOK_PER_WG + wave * TOK_PER_WAVE;
    const int g  = lane >> 4;              // half-wave group for WMMA layout
    const int ln = lane & 15;

    float bqv[8];
    #pragma unroll
    for (int nt = 0; nt < 8; ++nt) bqv[nt] = bq[nt * 16 + ln];

    for (int pass = 0; pass < 2; ++pass) {
        const int* qidx = (pass == 0) ? q : next_q;
        // pass0: (qe + sk + df + ar)/4
        // pass1: ((qe+sk+df+ar)/4 + sk+df+ar)/4 = (qe + 5*(sk+df+ar))/16
        const float w_qe = (pass == 0) ? 0.25f : 0.0625f;
        const float w_ot = (pass == 0) ? 0.25f : 0.3125f;

        // ---- gather + average: 32 lanes cooperate per token (4 dims each) ----
        for (int m = 0; m < TOK_PER_WAVE; ++m) {
            int t = tokBase + m;
            if (t >= n_tokens) t = n_tokens - 1;   // clamp (uniform across wave)
            const int qq = qidx[t];
            const int c4 = lane << 2;
            const float4 qe = *(const float4*)(ques_table + (long)qq * DDIM + c4);
            const float4 df = *(const float4*)(diff_table + (long)ques2diff[qq] * DDIM + c4);
            const float4 ar = *(const float4*)(area_table + (long)ques2area[qq] * DDIM + c4);
            float skx = 0.f, sky = 0.f, skz = 0.f, skw = 0.f, cnt = 0.f;
            #pragma unroll
            for (int kk = 0; kk < KSK; ++kk) {
                const int si = ques2skill[(long)qq * KSK + kk];
                const float msk = (si != pad_skill) ? 1.0f : 0.0f;  // branchless: EXEC stays full
                const float4 sr = *(const float4*)(skill_table + (long)si * DDIM + c4);
                skx += sr.x * msk; sky += sr.y * msk;
                skz += sr.z * msk; skw += sr.w * msk;
                cnt += msk;
            }
            const float inv = (cnt > 0.f) ? (1.0f / cnt) : 0.0f;
            float* p = stag + m * STRIDE + c4;
            p[0] = qe.x * w_qe + (skx * inv + df.x + ar.x) * w_ot;
            p[1] = qe.y * w_qe + (sky * inv + df.y + ar.y) * w_ot;
            p[2] = qe.z * w_qe + (skz * inv + df.z + ar.z) * w_ot;
            p[3] = qe.w * w_qe + (skw * inv + df.w + ar.w) * w_ot;
        }
        // wave-local LDS store->load ordering (DS ops in-order per wave; compiler waits)

        // ---- [16 x 128] @ Wq^T via V_WMMA_F32_16X16X4_F32 ----
        v8f acc[8];
        #pragma unroll
        for (int nt = 0; nt < 8; ++nt) { v8f z = {}; acc[nt] = z; }

        // A frag: lane l, vgpr j -> pre[l%16][k0 + j + 2*(l/16)]
        // B frag: lane l, vgpr j -> Wq[n0 + l%16][k0 + j + 2*(l/16)]   (B = Wq^T tile)
        const float* aBase = stag   + ln * STRIDE + 2 * g;
        const float* bBase = lds_wq + ln * STRIDE + 2 * g;
        for (int k0 = 0; k0 < DDIM; k0 += 4) {
            const v2f afrag = *(const v2f*)(aBase + k0);
            #pragma unroll
            for (int nt = 0; nt < 8; ++nt) {
                const v2f bfrag = *(const v2f*)(bBase + nt * 16 * STRIDE + k0);
                acc[nt] = __builtin_amdgcn_wmma_f32_16x16x4_f32(
                    false, afrag, false, bfrag, (short)0, acc[nt], false, false);
            }
        }

        // ---- epilogue: C layout M = r + 8*g, N = nt*16 + ln ----
        if (pass == 0) {
            #pragma unroll
            for (int r = 0; r < 8; ++r) {
                const int t = tokBase + r + 8 * g;
                if (t < n_tokens) {
                    const float af = (float)ans[t];
                    float* orow = out + (long)t * 384;
                    #pragma unroll
                    for (int nt = 0; nt < 8; ++nt) {
                        const float val = acc[nt][r] + bqv[nt];
                        orow[nt * 16 + ln]       = val * af;
                        orow[128 + nt * 16 + ln] = val * (1.0f - af);
                    }
                }
            }
        } else {
            #pragma unroll
            for (int r = 0; r < 8; ++r) {
                const int t = tokBase + r + 8 * g;
                if (t < n_tokens) {
                    float* orow = out + (long)t * 384 + 256;
                    #pragma unroll
                    for (int nt = 0; nt < 8; ++nt)
                        orow[nt * 16 + ln] = acc[nt][r] + bqv[nt];
                }
            }
        }
    }
}

extern "C" void kernel_launch(void* const* d_in, const int* in_sizes, int n_in,
                              void* d_out, int out_size, void* d_ws, size_t ws_size,
                              hipStream_t stream) {
    const int*   qv    = (const int*)d_in[0];
    const int*   av    = (const int*)d_in[1];
    const int*   nqv   = (const int*)d_in[2];
    const float* ques  = (const float*)d_in[3];
    const float* skill = (const float*)d_in[4];
    const float* diff  = (const float*)d_in[5];
    const float* area  = (const float*)d_in[6];
    const int*   q2s   = (const int*)d_in[7];
    const int*   q2d   = (const int*)d_in[8];
    const int*   q2a   = (const int*)d_in[9];
    const float* Wq    = (const float*)d_in[10];
    const float* bq    = (const float*)d_in[11];
    float* out = (float*)d_out;

    const int n_tok = in_sizes[0];                 // B*S
    const int NS    = in_sizes[4] / DDIM;          // skill rows
    const int pad   = NS - 1;

    const int grid = (n_tok + TOK_PER_WG - 1) / TOK_PER_WG;
    const size_t lds_bytes =
        (size_t)(DDIM * STRIDE + WAVES * TOK_PER_WAVE * STRIDE) * sizeof(float);

    h2kt_fused_wmma<<<grid, 128, lds_bytes, stream>>>(
        qv, av, nqv, ques, skill, diff, area, q2s, q2d, q2a, Wq, bq,
        out, n_tok, pad);
}